// Discriminator_30313879175350
// MI455X (gfx1250) — compile-verified
//
#include <hip/hip_runtime.h>

typedef __attribute__((ext_vector_type(2))) float v2f;
typedef __attribute__((ext_vector_type(8))) float v8f;

#define M_NODES (256 * 4096)   // B * N
#define NPT     4096           // nodes per tree
#define LVLS    6

__device__ inline void lds_fence() {
  // Wave-local LDS write->read ordering (CDNA5 split counters).
  asm volatile("s_wait_dscnt 0" ::: "memory");
}

__device__ inline v8f zero8() { v8f z = {}; return z; }

// One wave computes out[16][32] = act[16][K] @ W[K][32] with f32 WMMA 16x16x4.
// actL: LDS, row-major, row stride K.  W: global, row-major K x 32.
// c0 holds cols 0..15, c1 holds cols 16..31; element (j): row m = j + 8*(lane>=16),
// col n = lane&15  (per ISA 16x16 f32 C/D layout).
template <int K>
__device__ inline void wave_gemm(const float* actL, const float* W,
                                 int lane, v8f& c0, v8f& c1) {
  const int n    = lane & 15;          // A-row index == B/D column index for this lane
  const int koff = (lane >> 4) << 1;   // 0 or 2
#pragma unroll
  for (int k0 = 0; k0 < K; k0 += 4) {
    const int ka = k0 + koff;
    v2f a;  a.x  = actL[n * K + ka];          a.y  = actL[n * K + ka + 1];
    v2f b0; b0.x = W[ka * 32 + n];            b0.y = W[(ka + 1) * 32 + n];
    v2f b1; b1.x = W[ka * 32 + 16 + n];       b1.y = W[(ka + 1) * 32 + 16 + n];
    c0 = __builtin_amdgcn_wmma_f32_16x16x4_f32(false, a, false, b0, (short)0, c0, false, false);
    c1 = __builtin_amdgcn_wmma_f32_16x16x4_f32(false, a, false, b1, (short)0, c1, false, false);
  }
}

__device__ inline float pair_elem(long node, int k, const int* tok_a,
                                  const int* tok_b, const float* emb) {
  const int tok = (k < 16) ? tok_a[node] : tok_b[node];
  return emb[tok * 16 + (k & 15)];
}

// ---------------- init: h for height-0 nodes (leaf / ptr), 0 elsewhere ----------
__global__ __launch_bounds__(128) void k_init(
    const int* kind, const int* tok_a, const int* tok_b, const int* ptr_time,
    const float* first_notes, const float* lstm_out, const float* emb,
    const float* leaf_w1, const float* leaf_b1,
    const float* leaf_w2, const float* leaf_b2,
    const float* ptr_w, const float* ptr_b, float* hbuf) {
  __shared__ float sAct[4][16 * 128];
  __shared__ int   sKind[4][16];
  const int  wave = threadIdx.x >> 5;
  const int  lane = threadIdx.x & 31;
  const int  col  = lane & 15;
  const int  hi   = lane >> 4;
  const long base = ((long)blockIdx.x * 4 + wave) * 16;
  float* actL = sAct[wave];

  if (lane < 16) sKind[wave][lane] = kind[base + lane];

  // stage pair (16 x 32): [emb_a | emb_b]
  for (int e = lane; e < 16 * 32; e += 32) {
    const int r = e >> 5, k = e & 31;
    actL[r * 32 + k] = pair_elem(base + r, k, tok_a, tok_b, emb);
  }
  lds_fence();

  // leaf stage 1: pair @ leaf_w1 + b1 (no relu in reference)
  v8f c0 = zero8(), c1 = zero8();
  wave_gemm<32>(actL, leaf_w1, lane, c0, c1);
  const float b1lo = leaf_b1[col], b1hi = leaf_b1[col + 16];
  lds_fence();
#pragma unroll
  for (int j = 0; j < 8; ++j) {
    const int m = j + 8 * hi;
    actL[m * 32 + col]      = c0[j] + b1lo;
    actL[m * 32 + col + 16] = c1[j] + b1hi;
  }
  lds_fence();
  // leaf stage 2
  c0 = zero8(); c1 = zero8();
  wave_gemm<32>(actL, leaf_w2, lane, c0, c1);
  float lf0[8], lf1[8];
  const float b2lo = leaf_b2[col], b2hi = leaf_b2[col + 16];
#pragma unroll
  for (int j = 0; j < 8; ++j) { lf0[j] = c0[j] + b2lo; lf1[j] = c1[j] + b2hi; }
  lds_fence();

  // stage ptr_in (16 x 128): [first_notes(64) | lstm_out[tree, ptr_time](64)]
  for (int e = lane; e < 16 * 128; e += 32) {
    const int  r = e >> 7, k = e & 127;
    const long node = base + r;
    const int  tree = (int)(node >> 12);  // node / NPT
    float v;
    if (k < 64) v = first_notes[tree * 64 + k];
    else        v = lstm_out[((long)tree * 128 + ptr_time[node]) * 64 + (k - 64)];
    actL[r * 128 + k] = v;
  }
  lds_fence();
  c0 = zero8(); c1 = zero8();
  wave_gemm<128>(actL, ptr_w, lane, c0, c1);
  const float pblo = ptr_b[col], pbhi = ptr_b[col + 16];

  // select by kind; kind==2 (internal) gets 0 (csum accumulator init)
#pragma unroll
  for (int j = 0; j < 8; ++j) {
    const int  m  = j + 8 * hi;
    const int  kd = sKind[wave][m];
    const long node = base + m;
    const float v0 = (kd == 0) ? lf0[j] : (kd == 1) ? (c0[j] + pblo) : 0.0f;
    const float v1 = (kd == 0) ? lf1[j] : (kd == 1) ? (c1[j] + pbhi) : 0.0f;
    hbuf[node * 32 + col]      = v0;
    hbuf[node * 32 + col + 16] = v1;
  }
}

// ------------- scatter: each child adds its final h into its parent's slot ------
__global__ __launch_bounds__(256) void k_scatter(const int* parent,
                                                 const int* height,
                                                 float* hbuf, int lvl) {
  const long t = (long)blockIdx.x * blockDim.x + threadIdx.x;  // [0, M*8)
  const long node = t >> 3;
  if (node >= M_NODES) return;
  const int p = parent[node];
  if (p == (int)node) return;          // root self-loop contributes 0
  if (height[p] != lvl) return;        // child fires once, at parent's level
  const int cg = (int)(t & 7) * 4;
  const float4 v = *(const float4*)(hbuf + node * 32 + cg);
  float* dst = hbuf + (long)p * 32 + cg;
  atomicAdd(dst + 0, v.x);
  atomicAdd(dst + 1, v.y);
  atomicAdd(dst + 2, v.z);
  atomicAdd(dst + 3, v.w);
}

// ------------- node: h = relu([pair | csum] @ node_w + b) for height==lvl -------
__global__ __launch_bounds__(128) void k_node(
    const int* height, const int* tok_a, const int* tok_b, const float* emb,
    const float* node_w, const float* node_b, float* hbuf, int lvl) {
  __shared__ float sAct[4][16 * 64];
  __shared__ int   sH[4][16];
  const int  wave = threadIdx.x >> 5;
  const int  lane = threadIdx.x & 31;
  const int  col  = lane & 15;
  const int  hi   = lane >> 4;
  const long base = ((long)blockIdx.x * 4 + wave) * 16;

  const int hrow = (lane < 16) ? height[base + lane] : -1;
  if (!__any(hrow == lvl)) return;     // wave-uniform skip
  if (lane < 16) sH[wave][lane] = hrow;

  float* actL = sAct[wave];
  for (int e = lane; e < 16 * 64; e += 32) {
    const int  r = e >> 6, k = e & 63;
    const long node = base + r;
    actL[r * 64 + k] = (k < 32) ? pair_elem(node, k, tok_a, tok_b, emb)
                                : hbuf[node * 32 + (k - 32)];  // csum in place
  }
  lds_fence();
  v8f c0 = zero8(), c1 = zero8();
  wave_gemm<64>(actL, node_w, lane, c0, c1);
  const float blo = node_b[col], bhi = node_b[col + 16];
#pragma unroll
  for (int j = 0; j < 8; ++j) {
    const int m = j + 8 * hi;
    if (sH[wave][m] == lvl) {
      const long node = base + m;
      hbuf[node * 32 + col]      = fmaxf(c0[j] + blo, 0.0f);
      hbuf[node * 32 + col + 16] = fmaxf(c1[j] + bhi, 0.0f);
    }
  }
}

// ------------- head: 256 roots -> ff1 -> ff2 -> tail dot ------------------------
__global__ __launch_bounds__(512) void k_head(
    const float* hbuf, const float* ff_w1, const float* ff_b1,
    const float* ff_w2, const float* ff_b2, const float* tail_w,
    const float* tail_b, float* out) {
  __shared__ float sAct[16][16 * 32];
  const int wave = threadIdx.x >> 5;
  const int lane = threadIdx.x & 31;
  const int col  = lane & 15;
  const int hi   = lane >> 4;
  const int base = wave * 16;          // 16 waves x 16 trees = 256 roots
  float* actL = sAct[wave];

  for (int e = lane; e < 16 * 32; e += 32) {
    const int r = e >> 5, k = e & 31;
    actL[e] = hbuf[(long)(base + r) * NPT * 32 + k];  // root = node 0 of tree
  }
  lds_fence();
  v8f c0 = zero8(), c1 = zero8();
  wave_gemm<32>(actL, ff_w1, lane, c0, c1);
  lds_fence();
#pragma unroll
  for (int j = 0; j < 8; ++j) {
    const int m = j + 8 * hi;
    actL[m * 32 + col]      = c0[j] + ff_b1[col];
    actL[m * 32 + col + 16] = c1[j] + ff_b1[col + 16];
  }
  lds_fence();
  c0 = zero8(); c1 = zero8();
  wave_gemm<32>(actL, ff_w2, lane, c0, c1);
  lds_fence();
#pragma unroll
  for (int j = 0; j < 8; ++j) {
    const int m = j + 8 * hi;
    actL[m * 32 + col]      = c0[j] + ff_b2[col];
    actL[m * 32 + col + 16] = c1[j] + ff_b2[col + 16];
  }
  lds_fence();
  if (lane < 16) {
    float acc = tail_b[0];
#pragma unroll
    for (int k = 0; k < 32; ++k) acc += actL[lane * 32 + k] * tail_w[k];
    out[base + lane] = acc;
  }
}

extern "C" void kernel_launch(void* const* d_in, const int* in_sizes, int n_in,
                              void* d_out, int out_size, void* d_ws, size_t ws_size,
                              hipStream_t stream) {
  const int*   kind        = (const int*)d_in[0];
  const int*   height      = (const int*)d_in[1];
  const int*   parent      = (const int*)d_in[2];
  const int*   tok_a       = (const int*)d_in[3];
  const int*   tok_b       = (const int*)d_in[4];
  const int*   ptr_time    = (const int*)d_in[5];
  const float* first_notes = (const float*)d_in[6];
  const float* lstm_out    = (const float*)d_in[7];
  const float* emb         = (const float*)d_in[8];
  const float* leaf_w1     = (const float*)d_in[9];
  const float* leaf_b1     = (const float*)d_in[10];
  const float* leaf_w2     = (const float*)d_in[11];
  const float* leaf_b2     = (const float*)d_in[12];
  const float* node_w      = (const float*)d_in[13];
  const float* node_b      = (const float*)d_in[14];
  const float* ptr_w       = (const float*)d_in[15];
  const float* ptr_b       = (const float*)d_in[16];
  const float* ff_w1       = (const float*)d_in[17];
  const float* ff_b1       = (const float*)d_in[18];
  const float* ff_w2       = (const float*)d_in[19];
  const float* ff_b2       = (const float*)d_in[20];
  const float* tail_w      = (const float*)d_in[21];
  const float* tail_b      = (const float*)d_in[22];
  float* hbuf = (float*)d_ws;  // M * 32 floats: h / csum shared in place

  const int tileBlocks = (M_NODES / 16) / 4;           // 16384 (4 waves/block)
  k_init<<<tileBlocks, 128, 0, stream>>>(kind, tok_a, tok_b, ptr_time,
                                         first_notes, lstm_out, emb,
                                         leaf_w1, leaf_b1, leaf_w2, leaf_b2,
                                         ptr_w, ptr_b, hbuf);
  const int scatBlocks = (M_NODES * 8) / 256;          // 32768
  for (int lvl = 1; lvl <= LVLS; ++lvl) {
    k_scatter<<<scatBlocks, 256, 0, stream>>>(parent, height, hbuf, lvl);
    k_node<<<tileBlocks, 128, 0, stream>>>(height, tok_a, tok_b, emb,
                                           node_w, node_b, hbuf, lvl);
  }
  k_head<<<1, 512, 0, stream>>>(hbuf, ff_w1, ff_b1, ff_w2, ff_b2,
                                tail_w, tail_b, (float*)d_out);
}